// SamplerLayer_40226663694910
// MI455X (gfx1250) — compile-verified
//
#include <hip/hip_runtime.h>
#include <hip/hip_bf16.h>
#include <math.h>

// Problem geometry (fixed by the harness): B=256 rows, V=128000 vocab.
#define V_DIM 128000
#define B_DIM 256
#define SPLIT 5                        // chunks per row
#define CHUNK (V_DIM / SPLIT)          // 25600 elements per chunk
#define TPB 256                        // 8 wave32 waves
#define ELEMS_PER_ITER (TPB * 4)       // 1024 elements (float4 per lane)
#define ITERS (CHUNK / ELEMS_PER_ITER) // 25, exact
#define STAGES 4                       // async pipeline depth (LDS: 32 KB)

// CDNA5 async copy: each lane DMAs 16 bytes global -> LDS, tracked by ASYNCcnt.
// VDST operand = VGPR holding LDS byte offset (HW adds wave LDS_BASE).
// th:TH_LOAD_NT -- single-use 262MB stream > 192MB L2, don't cache it.
__device__ __forceinline__ void async_copy_b128(unsigned lds_off, const void* gaddr) {
    asm volatile("global_load_async_to_lds_b128 %0, %1, off th:TH_LOAD_NT"
                 :: "v"(lds_off), "v"(gaddr)
                 : "memory");
}

__global__ __launch_bounds__(TPB)
void sampler_partial_argmax(const float* __restrict__ logits,
                            const float* __restrict__ temperature,
                            const float* __restrict__ noise,
                            float* __restrict__ wsScore,
                            int*   __restrict__ wsIdx) {
    __shared__ float4 bufL[STAGES][TPB];   // 16 KB
    __shared__ float4 bufN[STAGES][TPB];   // 16 KB
    __shared__ float  swS[TPB / 32];
    __shared__ int    swI[TPB / 32];

    const int tid       = threadIdx.x;
    const int chunk     = blockIdx.x;
    const int row       = blockIdx.y;
    const int chunkBase = chunk * CHUNK;

    const float4* gL = (const float4*)(logits + (size_t)row * V_DIM + chunkBase) + tid;
    const float4* gN = (const float4*)(noise  + (size_t)row * V_DIM + chunkBase) + tid;

    // argmax( l/T - ln(max(n,1e-10)) ) == argmax( l * (1/(T*ln2)) - log2(max(n,1e-10)) )
    // v_rcp_f32 is plenty accurate for a sampler scale; avoids the IEEE div sequence.
    const float k = __builtin_amdgcn_rcpf(temperature[row]) * 1.44269504088896340736f;

    float bs = -INFINITY;
    int   bi = 0;

    // Prime STAGES-1 tiles of the async pipeline.
    #pragma unroll
    for (int s = 0; s < STAGES - 1; ++s) {
        async_copy_b128((unsigned)(size_t)&bufL[s][tid], gL + s * TPB);
        async_copy_b128((unsigned)(size_t)&bufN[s][tid], gN + s * TPB);
    }

    // Lane-private LDS slots: each lane consumes exactly the bytes its own async
    // op wrote, so ASYNCcnt ordering alone suffices -- no barriers in the loop.
    auto process = [&](int it) {
        const int cur = it & (STAGES - 1);
        const float4 l = bufL[cur][tid];
        const float4 n = bufN[cur][tid];
        const int base = chunkBase + it * ELEMS_PER_ITER + tid * 4;

        const float s0 = l.x * k - __log2f(fmaxf(n.x, 1e-10f));
        const float s1 = l.y * k - __log2f(fmaxf(n.y, 1e-10f));
        const float s2 = l.z * k - __log2f(fmaxf(n.z, 1e-10f));
        const float s3 = l.w * k - __log2f(fmaxf(n.w, 1e-10f));

        // Strict '>' keeps the earliest index (ascending walk) like jnp.argmax.
        if (s0 > bs) { bs = s0; bi = base;     }
        if (s1 > bs) { bs = s1; bi = base + 1; }
        if (s2 > bs) { bs = s2; bi = base + 2; }
        if (s3 > bs) { bs = s3; bi = base + 3; }
    };

    // Main loop: issue tile it+3, wait until only those 3 tiles (6 ops) remain.
    for (int it = 0; it < ITERS - (STAGES - 1); ++it) {
        const int nxt = (it + STAGES - 1) & (STAGES - 1);
        async_copy_b128((unsigned)(size_t)&bufL[nxt][tid], gL + (it + STAGES - 1) * TPB);
        async_copy_b128((unsigned)(size_t)&bufN[nxt][tid], gN + (it + STAGES - 1) * TPB);
        asm volatile("s_wait_asynccnt 0x6" ::: "memory");
        process(it);
    }
    // Drain epilogue: 3 tiles left in flight, then 2, then 1.
    asm volatile("s_wait_asynccnt 0x4" ::: "memory");
    process(ITERS - 3);
    asm volatile("s_wait_asynccnt 0x2" ::: "memory");
    process(ITERS - 2);
    asm volatile("s_wait_asynccnt 0x0" ::: "memory");
    process(ITERS - 1);

    // Wave32 butterfly reduction (tie -> smaller index).
    #pragma unroll
    for (int m = 16; m > 0; m >>= 1) {
        const float os = __shfl_xor(bs, m, 32);
        const int   oi = __shfl_xor(bi, m, 32);
        if (os > bs || (os == bs && oi < bi)) { bs = os; bi = oi; }
    }

    const int wave = tid >> 5;
    const int lane = tid & 31;
    if (lane == 0) { swS[wave] = bs; swI[wave] = bi; }
    __syncthreads();

    if (tid == 0) {
        #pragma unroll
        for (int w = 1; w < TPB / 32; ++w) {
            if (swS[w] > bs || (swS[w] == bs && swI[w] < bi)) { bs = swS[w]; bi = swI[w]; }
        }
        wsScore[row * SPLIT + chunk] = bs;
        wsIdx  [row * SPLIT + chunk] = bi;
    }
}

__global__ __launch_bounds__(TPB)
void sampler_final_argmax(const float* __restrict__ wsScore,
                          const int*   __restrict__ wsIdx,
                          int*         __restrict__ out) {
    const int row = threadIdx.x;
    if (row < B_DIM) {
        float bs = wsScore[row * SPLIT];
        int   bi = wsIdx  [row * SPLIT];
        #pragma unroll
        for (int p = 1; p < SPLIT; ++p) {
            const float s = wsScore[row * SPLIT + p];
            const int   i = wsIdx  [row * SPLIT + p];
            if (s > bs || (s == bs && i < bi)) { bs = s; bi = i; }
        }
        out[row] = bi;
    }
}

extern "C" void kernel_launch(void* const* d_in, const int* in_sizes, int n_in,
                              void* d_out, int out_size, void* d_ws, size_t ws_size,
                              hipStream_t stream) {
    const float* logits      = (const float*)d_in[0];
    const float* temperature = (const float*)d_in[1];
    const float* noise       = (const float*)d_in[2];

    float* wsScore = (float*)d_ws;                                        // B*SPLIT floats
    int*   wsIdx   = (int*)((char*)d_ws + sizeof(float) * B_DIM * SPLIT); // B*SPLIT ints

    dim3 grid(SPLIT, B_DIM);
    sampler_partial_argmax<<<grid, TPB, 0, stream>>>(logits, temperature, noise,
                                                     wsScore, wsIdx);
    sampler_final_argmax<<<1, TPB, 0, stream>>>(wsScore, wsIdx, (int*)d_out);
}